// MMD_60790967107815
// MI455X (gfx1250) — compile-verified
//
#include <hip/hip_runtime.h>

typedef __attribute__((ext_vector_type(2))) float v2f;
typedef __attribute__((ext_vector_type(8))) float v8f;

#define BATCH   4096
#define NROWS   8192
#define NTILES  512          // NROWS / 16
#define P1_BY   16           // grid.y of pass1
#define P1_BLOCKS (NTILES * P1_BY)

__device__ __forceinline__ const float* row_ptr(const float* __restrict__ src,
                                                const float* __restrict__ tgt, int r) {
    return (r < BATCH) ? (src + (size_t)r * 16) : (tgt + (size_t)(r - BATCH) * 16);
}

// Pass 0: per-row squared norms, analytic sum(l2) -> bandwidth -> exp2 scale.
__global__ void mmd_pass0(const float* __restrict__ src, const float* __restrict__ tgt,
                          float* __restrict__ ws_sq, float* __restrict__ ws_scale) {
    __shared__ float lvec[256 * 16];
    __shared__ float lsq[256];
    const int t = threadIdx.x;
    float vec[16];
#pragma unroll
    for (int c = 0; c < 16; c++) vec[c] = 0.f;
    float ssq = 0.f;
    for (int r = t; r < NROWS; r += 256) {
        const float* p = row_ptr(src, tgt, r);
        float sq = 0.f;
#pragma unroll
        for (int c = 0; c < 16; c++) { float x = p[c]; sq += x * x; vec[c] += x; }
        ws_sq[r] = sq;
        ssq += sq;
    }
#pragma unroll
    for (int c = 0; c < 16; c++) lvec[t * 16 + c] = vec[c];
    lsq[t] = ssq;
    __syncthreads();
    for (int s = 128; s > 0; s >>= 1) {
        if (t < s) {
#pragma unroll
            for (int c = 0; c < 16; c++) lvec[t * 16 + c] += lvec[(t + s) * 16 + c];
            lsq[t] += lsq[t + s];
        }
        __syncthreads();
    }
    if (t == 0) {
        float normsum = 0.f;
#pragma unroll
        for (int c = 0; c < 16; c++) { float v = lvec[c]; normsum += v * v; }
        // sum(l2) = 2n*sum(sq) - 2*||sum(x)||^2   (clamp is a no-op mathematically)
        float sumL2 = 2.f * (float)NROWS * lsq[0] - 2.f * normsum;
        double denom = (double)NROWS * (double)NROWS - (double)NROWS;
        float bw = (float)((double)sumL2 / denom) * 0.25f;   // / KERNEL_MUL^(NUM/2) = /4
        ws_scale[0] = 1.4426950408889634f / bw;              // log2(e)/bw
    }
}

// Pass 1: tiled Gram via V_WMMA_F32_16X16X4_F32, 5-kernel RBF sum, signed-weighted.
__global__ void __launch_bounds__(256)
mmd_pass1(const float* __restrict__ src, const float* __restrict__ tgt,
          const float* __restrict__ ws_sq, const float* __restrict__ ws_scale,
          float* __restrict__ partials) {
    __shared__ float red[256];
    const int tid  = threadIdx.x;
    const int lane = tid & 31;
    const int wv   = tid >> 5;
    const int I    = blockIdx.x;
    const int m    = lane & 15;
    const int h    = lane >> 4;                  // K-half selector (wave32 WMMA layout)
    const float s0 = ws_scale[0];

    // A tile (row-block I), loaded once per wave: lane holds float2 of its row.
    const float* ap = row_ptr(src, tgt, I * 16 + m);
    const v2f a0 = *(const v2f*)(ap + 0  + 2 * h);
    const v2f a1 = *(const v2f*)(ap + 4  + 2 * h);
    const v2f a2 = *(const v2f*)(ap + 8  + 2 * h);
    const v2f a3 = *(const v2f*)(ap + 12 + 2 * h);

    float sqA[8];
#pragma unroll
    for (int v = 0; v < 8; v++) sqA[v] = ws_sq[I * 16 + v + 8 * h];

    const float sI = (I < NTILES / 2) ? 1.f : -1.f;

    float acc = 0.f;
#pragma unroll
    for (int t = 0; t < 4; t++) {
        const int J = blockIdx.y * 32 + wv * 4 + t;   // wave-uniform
        if (J < I) continue;                          // symmetry: upper triangle only
        const float* bp = row_ptr(src, tgt, J * 16 + m);
        const v2f b0 = *(const v2f*)(bp + 0  + 2 * h);
        const v2f b1 = *(const v2f*)(bp + 4  + 2 * h);
        const v2f b2 = *(const v2f*)(bp + 8  + 2 * h);
        const v2f b3 = *(const v2f*)(bp + 12 + 2 * h);

        v8f c = {0.f, 0.f, 0.f, 0.f, 0.f, 0.f, 0.f, 0.f};
        c = __builtin_amdgcn_wmma_f32_16x16x4_f32(false, a0, false, b0, (short)0, c, false, false);
        c = __builtin_amdgcn_wmma_f32_16x16x4_f32(false, a1, false, b1, (short)0, c, false, false);
        c = __builtin_amdgcn_wmma_f32_16x16x4_f32(false, a2, false, b2, (short)0, c, false, false);
        c = __builtin_amdgcn_wmma_f32_16x16x4_f32(false, a3, false, b3, (short)0, c, false, false);

        const float sJ  = (J < NTILES / 2) ? 1.f : -1.f;
        const float w   = sI * sJ * ((J > I) ? 2.f : 1.f);
        const float sqB = ws_sq[J * 16 + m];
        float tsum = 0.f;
#pragma unroll
        for (int v = 0; v < 8; v++) {
            float l2 = sqA[v] + sqB - 2.f * c[v];
            l2 = fmaxf(l2, 0.f);
            const float u = -l2 * s0;                 // exp(-l2/bw_i) = exp2(u * 2^-i)
            tsum += __builtin_amdgcn_exp2f(u)
                  + __builtin_amdgcn_exp2f(u * 0.5f)
                  + __builtin_amdgcn_exp2f(u * 0.25f)
                  + __builtin_amdgcn_exp2f(u * 0.125f)
                  + __builtin_amdgcn_exp2f(u * 0.0625f);
        }
        acc += w * tsum;
    }

    red[tid] = acc;
    __syncthreads();
    for (int s = 128; s > 0; s >>= 1) {
        if (tid < s) red[tid] += red[tid + s];
        __syncthreads();
    }
    if (tid == 0) partials[blockIdx.x * P1_BY + blockIdx.y] = red[0];
}

// Pass 2: deterministic fixed-order reduction of block partials -> loss.
__global__ void mmd_pass2(const float* __restrict__ partials, float* __restrict__ out) {
    __shared__ float red[256];
    const int t = threadIdx.x;
    float s = 0.f;
    for (int i = t; i < P1_BLOCKS; i += 256) s += partials[i];
    red[t] = s;
    __syncthreads();
    for (int st = 128; st > 0; st >>= 1) {
        if (t < st) red[t] += red[t + st];
        __syncthreads();
    }
    if (t == 0) out[0] = red[0] / ((float)BATCH * (float)BATCH);
}

extern "C" void kernel_launch(void* const* d_in, const int* in_sizes, int n_in,
                              void* d_out, int out_size, void* d_ws, size_t ws_size,
                              hipStream_t stream) {
    const float* src = (const float*)d_in[0];
    const float* tgt = (const float*)d_in[1];
    float* ws        = (float*)d_ws;
    float* ws_sq     = ws;            // 8192 floats: per-row squared norms
    float* ws_scale  = ws + NROWS;    // 8 floats (1 used): log2(e)/bandwidth
    float* partials  = ws + NROWS + 8;// 8192 floats: per-block partial sums

    mmd_pass0<<<1, 256, 0, stream>>>(src, tgt, ws_sq, ws_scale);
    mmd_pass1<<<dim3(NTILES, P1_BY), 256, 0, stream>>>(src, tgt, ws_sq, ws_scale, partials);
    mmd_pass2<<<1, 256, 0, stream>>>(partials, (float*)d_out);
}